// MicroTopoBrain_18769007084203
// MI455X (gfx1250) — compile-verified
//
#include <hip/hip_runtime.h>
#include <hip/hip_bf16.h>
#include <math.h>

// ---------------------------------------------------------------------------
// MicroTopoBrain on MI455X (gfx1250, wave32)
//  - Heavy GEMMs: V_WMMA_F32_16X16X4_F32 (exact f32; both GEMMs are HBM-bound,
//    AI = 2*M/4 = 32 FLOP/B at M=64, so precision conversion buys nothing).
//  - Readout streams W_read (262 MB) through LDS via double-buffered
//    global_load_async_to_lds_b128 (ASYNCcnt), fully coalesced 16B segments.
//  - Two output tiles per wave share one A fragment (halves L2 A-traffic).
// ---------------------------------------------------------------------------

typedef float v2f __attribute__((ext_vector_type(2)));
typedef float v8f __attribute__((ext_vector_type(8)));

#define B_   64
#define FIN_ 256
#define G_   64
#define N_   4096
#define D_   16
#define ED_  65536
#define C_   1000

__device__ __forceinline__ float sigmoidf_(float x) { return 1.0f / (1.0f + __expf(-x)); }

// ---------------------------------------------------------------------------
// Kernel 1: x_embed[b,e] = sum_k x[b,k]*W_embed[e,k] + b_embed[e]
// One wave per 16(b) x 32(e) tile pair; A fragment (x) reused across both
// e-tiles: per K-step = 3 x b64 loads -> 2 x WMMA.
// f32 A 16x4 layout: VGPR0 = K {0 | 2} across lane halves, VGPR1 = K {1 | 3}.
// ---------------------------------------------------------------------------
__global__ __launch_bounds__(32)
void embed_wmma(const float* __restrict__ x, const float* __restrict__ W,
                const float* __restrict__ bias, float* __restrict__ out)
{
    const int lane = threadIdx.x;
    const int half = lane >> 4;          // 0: K+0..1, 1: K+2..3
    const int l16  = lane & 15;
    const int e0 = blockIdx.x * 32;
    const int b0 = blockIdx.y * 16;

    const float* xrow  = x + (size_t)(b0 + l16) * FIN_;
    const float* wrow0 = W + (size_t)(e0 + l16) * FIN_;
    const float* wrow1 = W + (size_t)(e0 + 16 + l16) * FIN_;

    v8f acc0 = {}, acc1 = {};
#pragma unroll 4
    for (int k = 0; k < FIN_; k += 4) {
        v2f a  = *(const v2f*)(xrow  + k + 2 * half);
        v2f b0 = *(const v2f*)(wrow0 + k + 2 * half);
        v2f b1 = *(const v2f*)(wrow1 + k + 2 * half);
        acc0 = __builtin_amdgcn_wmma_f32_16x16x4_f32(false, a, false, b0,
                                                     (short)0, acc0, false, false);
        acc1 = __builtin_amdgcn_wmma_f32_16x16x4_f32(false, a, false, b1,
                                                     (short)0, acc1, false, false);
    }
    const float be0 = bias[e0 + l16];
    const float be1 = bias[e0 + 16 + l16];
#pragma unroll
    for (int i = 0; i < 8; ++i) {
        // C layout: VGPR i, lanes 0-15 -> row i, lanes 16-31 -> row i+8; col = lane&15
        const size_t rowoff = (size_t)(b0 + i + 8 * half) * ED_;
        out[rowoff + (e0 + l16)]      = acc0[i] + be0;
        out[rowoff + (e0 + 16 + l16)] = acc1[i] + be1;
    }
}

// ---------------------------------------------------------------------------
// Kernel 2: cell[b,d] = sum_e x_embed[b,e] * W_cell[d,e] + b_cell[d]
// ---------------------------------------------------------------------------
__global__ __launch_bounds__(256)
void cell_reduce(const float* __restrict__ xe, const float* __restrict__ Wc,
                 const float* __restrict__ bc, float* __restrict__ cell)
{
    const int b = blockIdx.x >> 4;
    const int d = blockIdx.x & 15;
    const float4* xr = (const float4*)(xe + (size_t)b * ED_);
    const float4* wr = (const float4*)(Wc + (size_t)d * ED_);

    float s = 0.0f;
    for (int e = threadIdx.x; e < ED_ / 4; e += 256) {
        float4 a = xr[e], w = wr[e];
        s += a.x * w.x + a.y * w.y + a.z * w.z + a.w * w.w;
    }
    __shared__ float red[256];
    red[threadIdx.x] = s;
    __syncthreads();
    for (int off = 128; off; off >>= 1) {
        if (threadIdx.x < off) red[threadIdx.x] += red[threadIdx.x + off];
        __syncthreads();
    }
    if (threadIdx.x == 0) cell[b * D_ + d] = red[0] + bc[d];
}

// ---------------------------------------------------------------------------
// Kernel 3: fused per-row pipeline (one thread per (b,n) row of 16):
//   sparse 4-neighbor sigmoid-normalized aggregation -> stress/excitation ->
//   regulator MLP (3->8->3, LN+tanh+sigmoid) -> slow + fast*gate swish ->
//   layernorm(D) -> + cell broadcast  ==> combined2
// ---------------------------------------------------------------------------
__global__ __launch_bounds__(256)
void neuro_rows(const float* __restrict__ xe, const float* __restrict__ adj_w,
                const float* __restrict__ Wslow, const float* __restrict__ Wfast,
                const float* __restrict__ rw1, const float* __restrict__ rb1,
                const float* __restrict__ rlnw, const float* __restrict__ rlnb,
                const float* __restrict__ rw2, const float* __restrict__ rb2,
                const float* __restrict__ lnw, const float* __restrict__ lnb,
                const float* __restrict__ cell, float* __restrict__ comb)
{
    __shared__ float sWs[256], sWf[256];
    __shared__ float sRw1[24], sRw2[24], sRb1[8], sLnw[8], sLnb[8];
    __shared__ float sRb2[3], sLw[16], sLb[16];

    const int tid = threadIdx.x;
    sWs[tid] = Wslow[tid];
    sWf[tid] = Wfast[tid];
    if (tid < 24) { sRw1[tid] = rw1[tid]; sRw2[tid] = rw2[tid]; }
    if (tid < 8)  { sRb1[tid] = rb1[tid]; sLnw[tid] = rlnw[tid]; sLnb[tid] = rlnb[tid]; }
    if (tid < 3)  { sRb2[tid] = rb2[tid]; }
    if (tid < 16) { sLw[tid] = lnw[tid]; sLb[tid] = lnb[tid]; }
    __syncthreads();

    // fatigue = ||W_slow||_F
    float fro = 0.0f;
    for (int i = 0; i < 256; ++i) fro += sWs[i] * sWs[i];
    fro = sqrtf(fro);

    const int r = blockIdx.x * 256 + tid;   // 0 .. B*N-1
    const int b = r >> 12;                  // / N_
    const int n = r & (N_ - 1);
    const int gr = n >> 6, gc = n & (G_ - 1);
    const float* xb = xe + (size_t)b * ED_;
    const float* arow = adj_w + (size_t)n * N_;

    // ---- sparse aggregation with normalized sigmoid weights ----
    int   mN[4]; float wN[4]; int cnt = 0;
    if (gr > 0)      mN[cnt++] = n - G_;
    if (gr < G_ - 1) mN[cnt++] = n + G_;
    if (gc > 0)      mN[cnt++] = n - 1;
    if (gc < G_ - 1) mN[cnt++] = n + 1;
    float wsum = 0.0f;
    for (int j = 0; j < cnt; ++j) { wN[j] = sigmoidf_(arow[mN[j]]); wsum += wN[j]; }
    wsum = fmaxf(wsum, 1e-6f);
    float xa[16];
#pragma unroll
    for (int d = 0; d < 16; ++d) xa[d] = 0.0f;
    for (int j = 0; j < cnt; ++j) {
        const float wn = wN[j] / wsum;
        const float4* p = (const float4*)(xb + (size_t)mN[j] * D_);
#pragma unroll
        for (int q = 0; q < 4; ++q) {
            float4 v = p[q];
            xa[4 * q + 0] += wn * v.x; xa[4 * q + 1] += wn * v.y;
            xa[4 * q + 2] += wn * v.z; xa[4 * q + 3] += wn * v.w;
        }
    }

    // ---- stress: |var(ddof=1) - 0.5| ----
    float sum = 0.0f, sq = 0.0f;
#pragma unroll
    for (int d = 0; d < 16; ++d) { sum += xa[d]; sq += xa[d] * xa[d]; }
    const float mu = sum * (1.0f / 16.0f);
    const float var1 = (sq - 16.0f * mu * mu) * (1.0f / 15.0f);
    const float stress = fabsf(var1 - 0.5f);

    // ---- h_raw = xf @ W_slow^T ; excitation ----
    float hr[16]; float exc = 0.0f;
#pragma unroll
    for (int d2 = 0; d2 < 16; ++d2) {
        float s = 0.0f;
#pragma unroll
        for (int d = 0; d < 16; ++d) s += xa[d] * sWs[d2 * 16 + d];
        hr[d2] = s;
        exc += fabsf(s);
    }
    exc *= (1.0f / 16.0f);

    // ---- regulator MLP ----
    const float st[3] = { stress, exc, fro };
    float h1[8]; float m8 = 0.0f;
#pragma unroll
    for (int j = 0; j < 8; ++j) {
        float s = sRb1[j];
#pragma unroll
        for (int i = 0; i < 3; ++i) s += st[i] * sRw1[j * 3 + i];
        h1[j] = s; m8 += s;
    }
    m8 *= 0.125f;
    float v8 = 0.0f;
#pragma unroll
    for (int j = 0; j < 8; ++j) { float t = h1[j] - m8; v8 += t * t; }
    v8 *= 0.125f;
    const float inv8 = rsqrtf(v8 + 1e-5f);
    float hh[8];
#pragma unroll
    for (int j = 0; j < 8; ++j) hh[j] = tanhf((h1[j] - m8) * inv8 * sLnw[j] + sLnb[j]);
    float ctrl[3];
#pragma unroll
    for (int i = 0; i < 3; ++i) {
        float s = sRb2[i];
#pragma unroll
        for (int j = 0; j < 8; ++j) s += hh[j] * sRw2[i * 8 + j];
        ctrl[i] = sigmoidf_(s);
    }
    const float sens = ctrl[1], gate = ctrl[2];
    const float beta = 0.5f + 2.0f * sens;

    // ---- combined = slow + fast*gate; swish(beta); layernorm(D); +cell ----
    float o[16]; float msum = 0.0f;
#pragma unroll
    for (int d2 = 0; d2 < 16; ++d2) {
        float f = 0.0f;
#pragma unroll
        for (int d = 0; d < 16; ++d) f += xa[d] * sWf[d2 * 16 + d];
        const float c = hr[d2] + f * gate;
        const float out = c * sigmoidf_(beta * c);
        o[d2] = out; msum += out;
    }
    msum *= (1.0f / 16.0f);
    float vsum = 0.0f;
#pragma unroll
    for (int d2 = 0; d2 < 16; ++d2) { float t = o[d2] - msum; vsum += t * t; }
    vsum *= (1.0f / 16.0f);
    const float inv = rsqrtf(vsum + 1e-5f);

    const float* cb = cell + b * D_;
    float4* dst = (float4*)(comb + (size_t)r * D_);
#pragma unroll
    for (int q = 0; q < 4; ++q) {
        float4 v;
        v.x = (o[4*q+0] - msum) * inv * sLw[4*q+0] + sLb[4*q+0] + cb[4*q+0];
        v.y = (o[4*q+1] - msum) * inv * sLw[4*q+1] + sLb[4*q+1] + cb[4*q+1];
        v.z = (o[4*q+2] - msum) * inv * sLw[4*q+2] + sLb[4*q+2] + cb[4*q+2];
        v.w = (o[4*q+3] - msum) * inv * sLw[4*q+3] + sLb[4*q+3] + cb[4*q+3];
        dst[q] = v;
    }
}

// ---------------------------------------------------------------------------
// Kernel 4: logits init with bias (atomic split-K accumulates onto this)
// ---------------------------------------------------------------------------
__global__ __launch_bounds__(256)
void init_logits(const float* __restrict__ b_read, float* __restrict__ logits)
{
    const int i = blockIdx.x * 256 + threadIdx.x;
    if (i < B_ * C_) logits[i] = b_read[i % C_];
}

// ---------------------------------------------------------------------------
// Kernel 5: logits[b,c] += sum_k combined2[b,k] * W_read[c,k]   (split-K WMMA)
// Grid: (32 c-pair-tiles, 4 b-tiles, 64 k-chunks of 1024). One wave/block.
//
// B tiles (32 c-rows x 64 k, rows padded to 68 floats for bank-conflict-free
// ds_load_b64) are streamed into LDS with double-buffered
// global_load_async_to_lds_b128 (fully coalesced 16B segments, ASYNCcnt
// tracked, overlapped with WMMA on the other buffer). Single-wave workgroup:
// no barriers needed, hazards ordered by s_wait_asynccnt + program order.
// The single __shared__ array starts at LDS offset 0, so raw byte offsets
// are passed directly to the async instruction.
// ---------------------------------------------------------------------------
#define KT_    64                  // k per tile
#define NT_    16                  // tiles per 1024-k chunk
#define ROWF_  68                  // padded row stride (floats): 68 % 64 = 4
#define TILEF_ (32 * ROWF_)        // floats per buffer (2176)

__global__ __launch_bounds__(32)
void readout_wmma(const float* __restrict__ comb, const float* __restrict__ Wr,
                  float* __restrict__ logits)
{
    __shared__ float sB[2 * TILEF_];   // 17408 B, double-buffered B tiles

    const int lane = threadIdx.x;
    const int half = lane >> 4;
    const int l16  = lane & 15;
    const int cbase = blockIdx.x * 32;
    const int b0    = blockIdx.y * 16;
    const int k0    = blockIdx.z * 1024;

    const float* arow = comb + (size_t)(b0 + l16) * ED_;

    // ---- async prefetch of one B tile (32 rows x 64 k) into buffer `buf` ----
    auto prefetch = [&](int buf, int kk) {
#pragma unroll
        for (int q = 0; q < 16; ++q) {
            const int seg = q * 32 + lane;      // 512 x 16B segments
            const int row = seg >> 4;           // 0..31
            const int col = seg & 15;           // 16B units within row
            int rg = cbase + row;               // clamp OOB c-rows (masked at store)
            if (rg >= C_) rg = C_ - 1;
            const unsigned long long gaddr =
                (unsigned long long)(Wr + (size_t)rg * ED_ + kk + col * 4);
            const unsigned ldsoff =
                (unsigned)((buf * TILEF_ + row * ROWF_ + col * 4) * 4);
            asm volatile("global_load_async_to_lds_b128 %0, %1, off"
                         :: "v"(ldsoff), "v"(gaddr) : "memory");
        }
    };

    v8f acc0 = {}, acc1 = {};
    prefetch(0, k0);
#pragma unroll 1
    for (int t = 0; t < NT_; ++t) {
        if (t + 1 < NT_) {
            prefetch((t + 1) & 1, k0 + (t + 1) * KT_);
            asm volatile("s_wait_asynccnt 0x10" ::: "memory");  // tile t landed
        } else {
            asm volatile("s_wait_asynccnt 0x0" ::: "memory");
        }
        const int   kg = k0 + t * KT_;
        const float* bbase = &sB[(t & 1) * TILEF_];
#pragma unroll
        for (int ks = 0; ks < KT_; ks += 4) {
            v2f a  = *(const v2f*)(arow + kg + ks + 2 * half);
            v2f b0 = *(const v2f*)(bbase + l16 * ROWF_ + ks + 2 * half);
            v2f b1 = *(const v2f*)(bbase + (16 + l16) * ROWF_ + ks + 2 * half);
            acc0 = __builtin_amdgcn_wmma_f32_16x16x4_f32(false, a, false, b0,
                                                         (short)0, acc0, false, false);
            acc1 = __builtin_amdgcn_wmma_f32_16x16x4_f32(false, a, false, b1,
                                                         (short)0, acc1, false, false);
        }
    }

    const int c0 = cbase + l16;
    const int c1 = cbase + 16 + l16;
#pragma unroll
    for (int i = 0; i < 8; ++i) {
        const size_t rowoff = (size_t)(b0 + i + 8 * half) * C_;
        if (c0 < C_) atomicAdd(&logits[rowoff + c0], acc0[i]);
        if (c1 < C_) atomicAdd(&logits[rowoff + c1], acc1[i]);
    }
}

// ---------------------------------------------------------------------------
// Launch
// ---------------------------------------------------------------------------
extern "C" void kernel_launch(void* const* d_in, const int* in_sizes, int n_in,
                              void* d_out, int out_size, void* d_ws, size_t ws_size,
                              hipStream_t stream)
{
    const float* x       = (const float*)d_in[0];
    const float* W_embed = (const float*)d_in[1];
    const float* b_embed = (const float*)d_in[2];
    const float* adj_w   = (const float*)d_in[3];
    const float* W_slow  = (const float*)d_in[4];
    const float* W_fast  = (const float*)d_in[5];
    const float* reg_w1  = (const float*)d_in[6];
    const float* reg_b1  = (const float*)d_in[7];
    const float* reg_lnw = (const float*)d_in[8];
    const float* reg_lnb = (const float*)d_in[9];
    const float* reg_w2  = (const float*)d_in[10];
    const float* reg_b2  = (const float*)d_in[11];
    const float* ln_w    = (const float*)d_in[12];
    const float* ln_b    = (const float*)d_in[13];
    const float* W_cell  = (const float*)d_in[14];
    const float* b_cell  = (const float*)d_in[15];
    const float* W_read  = (const float*)d_in[16];
    const float* b_read  = (const float*)d_in[17];
    float* logits = (float*)d_out;

    char* ws = (char*)d_ws;
    float* x_embed = (float*)(ws);                                        // 16 MB
    float* comb2   = (float*)(ws + (size_t)B_ * ED_ * sizeof(float));     // 16 MB
    float* cell    = (float*)(ws + (size_t)2 * B_ * ED_ * sizeof(float)); // 4 KB
    (void)in_sizes; (void)n_in; (void)out_size; (void)ws_size;

    // 1) embedding GEMM (WMMA f32, paired e-tiles)
    embed_wmma<<<dim3(ED_ / 32, B_ / 16), 32, 0, stream>>>(x, W_embed, b_embed, x_embed);

    // 2) cell = x_embed @ W_cell^T + b_cell
    cell_reduce<<<dim3(B_ * D_), 256, 0, stream>>>(x_embed, W_cell, b_cell, cell);

    // 3) fused aggregation + regulator + swish + LN + cell broadcast
    neuro_rows<<<dim3((B_ * N_) / 256), 256, 0, stream>>>(
        x_embed, adj_w, W_slow, W_fast, reg_w1, reg_b1, reg_lnw, reg_lnb,
        reg_w2, reg_b2, ln_w, ln_b, cell, comb2);

    // 4) logits = b_read (broadcast)
    init_logits<<<dim3((B_ * C_ + 255) / 256), 256, 0, stream>>>(b_read, logits);

    // 5) readout GEMM (split-K WMMA, async-LDS double-buffered B, f32 atomics)
    readout_wmma<<<dim3((C_ + 31) / 32, B_ / 16, 64), 32, 0, stream>>>(comb2, W_read, logits);
}